// FSMNKernel_16776142258544
// MI455X (gfx1250) — compile-verified
//
#include <hip/hip_runtime.h>

// ---------------------------------------------------------------------------
// FSMN memory block (strided dilated depthwise conv over time) for MI455X.
// HBM-bound (AI ~ 4 flop/byte). Strategy:
//   * Tensor Data Mover stages a [frames x channels] tile (incl. temporal
//     halo) Global->LDS asynchronously (TENSORcnt + workgroup barrier).
//   * 96-frame x 128-channel tiles: halo overhead 30% (vs 90% at 32 frames),
//     62.5 KB LDS -> ~5 blocks/WGP so DMA latency hides under occupancy.
//   * float2 per thread: ds_load_b64 taps, v_pk_fma_f32 math, 8B NT stores.
// ---------------------------------------------------------------------------

typedef __attribute__((ext_vector_type(4))) unsigned int v4u;
typedef __attribute__((ext_vector_type(8))) int          v8i;
typedef __attribute__((ext_vector_type(4))) int          v4i;
typedef __attribute__((ext_vector_type(2))) float        f2;

#define B_      16
#define T_      2000
#define D_      512
#define LORD    10
#define RORD    5
#define NTAP    (LORD + 1 + RORD)
#define LPAD    20           // LORD * l_stride
#define T_TILE  96
#define ROWS    (LPAD + T_TILE + RORD * 2 - 1)   // 125
#define DT      128          // channels per block (4 d-tiles cover D=512)
#define QFRM    (T_TILE / 4) // 24 frames per thread-quarter

__global__ __launch_bounds__(256)
void fsmn_tdm_kernel(const float* __restrict__ in,
                     const float* __restrict__ filt,
                     float* __restrict__ out)
{
    __shared__ float smem[ROWS * DT];   // 62.5 KB, rows of 512 B

    const int tid  = threadIdx.x;
    const int pair = tid & 63;          // float2 column: channels 2p, 2p+1
    const int qh   = tid >> 6;          // frame quarter 0..3
    const int dt   = blockIdx.x;        // 0..3 channel tile
    const int t0   = blockIdx.y * T_TILE;
    const int b    = blockIdx.z;
    const int c    = dt * DT + 2 * pair;   // global channel (even)

    const int halo_start = t0 - LPAD;      // global frame of LDS row 0
    const int lastf      = halo_start + ROWS - 1;

    // ---- zero out-of-range halo rows (column-private, no races) -----------
    int zf = -halo_start; if (zf < 0) zf = 0; if (zf > ROWS) zf = ROWS;
    if (qh == 0) {
        for (int r = 0; r < zf; ++r)
            *(f2*)&smem[r * DT + 2 * pair] = (f2){0.0f, 0.0f};
    }
    int zb = lastf - (T_ - 1); if (zb < 0) zb = 0; if (zb > ROWS) zb = ROWS;
    if (qh == 3) {
        for (int r = ROWS - zb; r < ROWS; ++r)
            *(f2*)&smem[r * DT + 2 * pair] = (f2){0.0f, 0.0f};
    }

    // ---- TDM async tensor load: [nvalid x 128] strided tile -> LDS --------
    const int vf0    = (halo_start < 0) ? 0 : halo_start;
    const int vf1    = (lastf >= T_) ? (T_ - 1) : lastf;
    const int nvalid = vf1 - vf0 + 1;

    if ((tid >> 5) == 0) {  // wave 0 issues the DMA (uniform branch)
        unsigned lds_addr = (unsigned)(size_t)(&smem[0])
                          + (unsigned)(vf0 - halo_start) * (DT * 4u);
        unsigned long long gaddr = (unsigned long long)(size_t)in
            + (((unsigned long long)((size_t)b * T_ + (size_t)vf0)) * D_
               + (unsigned long long)(dt * DT)) * 4ull;

        // D# group 0 (ISA 8.3): count=1 | lds_addr | global_addr[56:0] | type=2
        v4u g0;
        g0.x = 1u;
        g0.y = lds_addr;
        g0.z = (unsigned)(gaddr & 0xFFFFFFFFull);
        g0.w = (unsigned)((gaddr >> 32) & 0x1FFFFFFull) | (2u << 30);

        // D# group 1 (ISA 8.4): data_size=4B; tensor 128 x nvalid,
        // row stride 512 elements (full D); tile = 128 x nvalid.
        v8i g1;
        g1[0] = (int)(2u << 16);                 // data_size = 2 (4 bytes)
        g1[1] = (int)((DT & 0xFFFF) << 16);      // tensor_dim0 lo16 (=128)
        g1[2] = (int)((nvalid & 0xFFFF) << 16);  // dim0 hi16 | tensor_dim1 lo16
        g1[3] = (int)((DT & 0xFFFF) << 16);      // dim1 hi16 | tile_dim0 (=128)
        g1[4] = (int)(nvalid & 0xFFFF);          // tile_dim1 | tile_dim2=0
        g1[5] = (int)D_;                         // tensor_dim0_stride lo32 (=512)
        g1[6] = 0;                               // stride hi16 | dim1_stride lo16
        g1[7] = 0;
        v4i gz = {0, 0, 0, 0};
#if __clang_major__ >= 23
        v8i gz8 = {0, 0, 0, 0, 0, 0, 0, 0};
        __builtin_amdgcn_tensor_load_to_lds(g0, g1, gz, gz, gz8, 0);
#else
        __builtin_amdgcn_tensor_load_to_lds(g0, g1, gz, gz, 0);
#endif
    }
    __builtin_amdgcn_s_wait_tensorcnt(0);   // issuer drains TENSORcnt
    __syncthreads();                        // publish LDS tile + zeros

    // ---- per-thread filter taps (float2); fuse residual into center -------
    f2 fw[NTAP];
#pragma unroll
    for (int k = 0; k < NTAP; ++k) fw[k] = *(const f2*)&filt[k * D_ + c];
    fw[LORD].x += 1.0f;   // out += inputs (KERNEL_RES)
    fw[LORD].y += 1.0f;

    // ---- compute this quarter's frames from LDS, NT-store float2 ----------
    int iLim = T_ - t0; if (iLim > T_TILE) iLim = T_TILE;
    int iEnd = qh * QFRM + QFRM; if (iEnd > iLim) iEnd = iLim;
    for (int i = qh * QFRM; i < iEnd; ++i) {
        const float* col = &smem[i * DT + 2 * pair];
        f2 acc = (f2){0.0f, 0.0f};
#pragma unroll
        for (int k = 0; k < LORD; ++k)                  // left taps: rows i+2k
            acc = *(const f2*)&col[(2 * k) * DT] * fw[k] + acc;
        acc = *(const f2*)&col[LPAD * DT] * fw[LORD] + acc;   // center+residual
#pragma unroll
        for (int k = 0; k < RORD; ++k)                  // right taps: i+21+2k
            acc = *(const f2*)&col[(LPAD + 1 + 2 * k) * DT] * fw[LORD + 1 + k] + acc;
        __builtin_nontemporal_store(
            acc, (f2*)&out[((size_t)b * T_ + (size_t)(t0 + i)) * D_ + (size_t)c]);
    }
}

extern "C" void kernel_launch(void* const* d_in, const int* in_sizes, int n_in,
                              void* d_out, int out_size, void* d_ws, size_t ws_size,
                              hipStream_t stream) {
    (void)in_sizes; (void)n_in; (void)ws_size; (void)d_ws; (void)out_size;
    const float* in   = (const float*)d_in[0];   // (B, T, D) fp32
    const float* filt = (const float*)d_in[1];   // (16, D)   fp32
    float*       out  = (float*)d_out;           // (B, T, D) fp32

    dim3 grid(D_ / DT, (T_ + T_TILE - 1) / T_TILE, B_);  // (4, 21, 16)
    dim3 block(256);
    fsmn_tdm_kernel<<<grid, block, 0, stream>>>(in, filt, out);
}